// EMA_75316546502844
// MI455X (gfx1250) — compile-verified
//
#include <hip/hip_runtime.h>
#include <math.h>

// EMA scan: out[t, ch, e] = d_e * x[t, ch] + (1 - d_e) * out[t-1, ch, e],
// init = x[0].  x: (4096, 2048) f32 flat; out: (4096, 2048, 8) f32 flat.
//
// Chunked-scan decomposition with the intra-chunk expansion done as a
// lower-triangular matmul on V_WMMA_F32_16X16X4_F32 (full f32 precision).

typedef __attribute__((ext_vector_type(2))) float v2f;
typedef __attribute__((ext_vector_type(8))) float v8f;

#define T_LEN   4096
#define CH_LEN  2048   // 8 * 256
#define EMAS    8
#define CHUNK   16
#define NCHUNK  (T_LEN / CHUNK)   // 256
#define KITER   8                 // chunks per wave in pass 3

// branchless a^p for p in [0, 16] via binary decomposition
__device__ __forceinline__ float apow(int p, float a, float a2, float a4,
                                      float a8, float a16) {
    float r = (p & 1) ? a : 1.0f;
    r *= (p & 2) ? a2 : 1.0f;
    r *= (p & 4) ? a4 : 1.0f;
    r *= (p & 8) ? a8 : 1.0f;
    r *= (p & 16) ? a16 : 1.0f;
    return r;
}

// coefficient d * a^(t-s) for s<=t, else 0   (t-s in [0,15])
__device__ __forceinline__ float coefTS(int s, int t, float d, float a,
                                        float a2, float a4, float a8) {
    int p = t - s;
    float r = d;
    r *= (p & 1) ? a : 1.0f;
    r *= (p & 2) ? a2 : 1.0f;
    r *= (p & 4) ? a4 : 1.0f;
    r *= (p & 8) ? a8 : 1.0f;
    return (s <= t) ? r : 0.0f;
}

// ---------------------------------------------------------------------------
// Pass 1: per-chunk carries  c_k[ch,e] = sum_{j=0..15} a^(15-j) * d * x[t0+j,ch]
// One thread per (chunk, channel). Horner over j; 8 emas per thread so x is
// read exactly once.
// ---------------------------------------------------------------------------
__global__ void ema_carries_kernel(const float* __restrict__ x,
                                   const float* __restrict__ log_decay,
                                   float* __restrict__ carries) {
    int tid = blockIdx.x * blockDim.x + threadIdx.x;   // 0 .. 256*2048-1
    int ch = tid & (CH_LEN - 1);
    int k  = tid >> 11;

    const float* xp = x + (size_t)k * CHUNK * CH_LEN + ch;
    float xs[CHUNK];
#pragma unroll
    for (int j = 0; j < CHUNK; ++j) xs[j] = xp[(size_t)j * CH_LEN];

    float* cp = carries + ((size_t)k * CH_LEN + ch) * EMAS;
#pragma unroll
    for (int e = 0; e < EMAS; ++e) {
        float d = 1.0f / (1.0f + expf(-log_decay[e]));
        float a = 1.0f - d;
        float c = 0.0f;
#pragma unroll
        for (int j = 0; j < CHUNK; ++j) c = a * c + d * xs[j];
        cp[e] = c;
    }
}

// ---------------------------------------------------------------------------
// Pass 2: sequential scan over chunks, in place: buffer holds carries on
// input and the state ENTERING each chunk on output.
//   S_0 = x[0, ch];  S_{k+1} = a^16 * S_k + c_k
// One thread per (ch, e): 16384 chains of length 256.
// ---------------------------------------------------------------------------
__global__ void ema_chunkscan_kernel(const float* __restrict__ x,
                                     const float* __restrict__ log_decay,
                                     float* __restrict__ cs) {
    int tid = blockIdx.x * blockDim.x + threadIdx.x;   // 0 .. 16383
    int e  = tid & (EMAS - 1);
    int ch = tid >> 3;

    float d = 1.0f / (1.0f + expf(-log_decay[e]));
    float a = 1.0f - d;
    float a2 = a * a, a4 = a2 * a2, a8 = a4 * a4, a16 = a8 * a8;

    float S = x[ch];                     // init = x[0, ch]
    size_t idx = (size_t)ch * EMAS + e;
    const size_t stride = (size_t)CH_LEN * EMAS;
    for (int k = 0; k < NCHUNK; ++k) {
        float c = cs[idx];
        cs[idx] = S;                     // state entering chunk k
        S = a16 * S + c;
        idx += stride;
    }
}

// ---------------------------------------------------------------------------
// Pass 3: intra-chunk expansion as triangular matmul on WMMA f32 16x16x4.
// One wave per (chunk-group, 16-channel group); each wave runs KITER chunks
// so the decay-coefficient B fragments (constant across chunks/channels) are
// built once and held in registers.
//   D[ch, n] = sum_s A[ch,s] * B[s,n] + C[ch,n]
//   A[ch,s]  = x[t0+s, cb+ch]                      (16x16 via 4 K-steps)
//   B[s,n]   = (s<=t) ? d_e * a_e^(t-s) : 0,  n = (t-tb)*8 + e
//   C[ch,n]  = a_e^(t+1) * S[cb+ch, e]             (carry-in)
// A 16x4 layout: lane l: M = l%16, VGPR v: K = v + 2*(l>=16).
// B 4x16 layout (mirrored): lane l: N = l%16, VGPR v: K = v + 2*(l>=16).
// D 16x16 layout: lane l: N = l%16, VGPR v: M = v + 8*(l>=16).
// ---------------------------------------------------------------------------
__global__ void ema_wmma_kernel(const float* __restrict__ x,
                                const float* __restrict__ log_decay,
                                const float* __restrict__ states,
                                float* __restrict__ out) {
    int gw   = (blockIdx.x * blockDim.x + threadIdx.x) >> 5; // global wave
    int lane = threadIdx.x & 31;
    int kg = gw >> 7;         // chunk group   0..31
    int g  = gw & 127;        // channel group 0..127
    int cb = g * 16;

    int h  = lane >> 4;       // lane half (K/M offset selector)
    int n  = lane & 15;       // N (and A's M) index
    int e  = lane & 7;        // ema index inside N
    int tp = (lane >> 3) & 1; // time within the t-pair

    float d = 1.0f / (1.0f + expf(-log_decay[e]));
    float a = 1.0f - d;
    float a2 = a * a, a4 = a2 * a2, a8 = a4 * a4, a16 = a8 * a8;

    // Hoisted per-lane constants: B fragments + carry powers (chunk-invariant)
    v2f   bfr[8][4];
    float cpw[8];
#pragma unroll
    for (int tbi = 0; tbi < 8; ++tbi) {
        int t = 2 * tbi + tp;                    // per-lane output time
        cpw[tbi] = apow(t + 1, a, a2, a4, a8, a16);
#pragma unroll
        for (int j = 0; j < 4; ++j) {            // K chunk, s0 = 4j
            int s = 4 * j + 2 * h;
            bfr[tbi][j].x = coefTS(s,     t, d, a, a2, a4, a8);
            bfr[tbi][j].y = coefTS(s + 1, t, d, a, a2, a4, a8);
        }
    }

#pragma unroll 1
    for (int kk = 0; kk < KITER; ++kk) {
        int k  = kg * KITER + kk;
        int t0 = k * CHUNK;

        // A fragments: x tile, col = cb + n, rows t0 + (4j+2h) / (4j+2h+1)
        const float* xc = x + (size_t)t0 * CH_LEN + cb + n;
        v2f afr[4];
#pragma unroll
        for (int j = 0; j < 4; ++j) {
            afr[j].x = xc[(size_t)(4 * j + 2 * h) * CH_LEN];
            afr[j].y = xc[(size_t)(4 * j + 2 * h + 1) * CH_LEN];
        }

        // State fragment: S[ch = v + 8h, e]  (D-layout aligned)
        v8f sfr;
        const float* sp = states + (((size_t)k * CH_LEN + cb + 8 * h) * EMAS + e);
#pragma unroll
        for (int v = 0; v < 8; ++v) sfr[v] = sp[(size_t)v * EMAS];

#pragma unroll
        for (int tbi = 0; tbi < 8; ++tbi) {
            int t = 2 * tbi + tp;
            v8f acc;
#pragma unroll
            for (int v = 0; v < 8; ++v) acc[v] = sfr[v] * cpw[tbi];

#pragma unroll
            for (int j = 0; j < 4; ++j)
                acc = __builtin_amdgcn_wmma_f32_16x16x4_f32(
                    false, afr[j], false, bfr[tbi][j], (short)0, acc,
                    false, false);

            // out[(t0+t)*2048 + cb + v + 8h][e], streaming (never re-read)
            size_t ob = (((size_t)(t0 + t) * CH_LEN) + cb + 8 * h) * EMAS + e;
#pragma unroll
            for (int v = 0; v < 8; ++v)
                __builtin_nontemporal_store(acc[v],
                                            &out[ob + (size_t)v * EMAS]);
        }
    }
}

// ---------------------------------------------------------------------------
extern "C" void kernel_launch(void* const* d_in, const int* in_sizes, int n_in,
                              void* d_out, int out_size, void* d_ws, size_t ws_size,
                              hipStream_t stream) {
    const float* x  = (const float*)d_in[0];       // (4096, 8, 256) f32
    const float* ld = (const float*)d_in[1];       // (8,) f32
    float* out = (float*)d_out;                    // (4096, 8, 256, 8) f32
    float* cs  = (float*)d_ws;                     // carries -> states, 16 MB

    // Pass 1: 256 chunks * 2048 channels threads
    ema_carries_kernel<<<(NCHUNK * CH_LEN) / 256, 256, 0, stream>>>(x, ld, cs);
    // Pass 2: 2048 channels * 8 emas threads
    ema_chunkscan_kernel<<<(CH_LEN * EMAS) / 256, 256, 0, stream>>>(x, ld, cs);
    // Pass 3: (256/KITER) chunk groups * 128 channel groups waves, 8 waves/block
    ema_wmma_kernel<<<((NCHUNK / KITER) * 128) / 8, 256, 0, stream>>>(x, ld, cs, out);
}